// NnHalfKP_49555332661320
// MI455X (gfx1250) — compile-verified
//
#include <hip/hip_runtime.h>

typedef __attribute__((ext_vector_type(2))) float v2f;
typedef __attribute__((ext_vector_type(8))) float v8f;

#define B_SZ   8192
#define FT_INN 40960
#define FT_OUT 256
#define VFT    640
#define HID    512   // 2 * FT_OUT

static __device__ __forceinline__ void atomAddF(float* p, float v) {
  __hip_atomic_fetch_add(p, v, __ATOMIC_RELAXED, __HIP_MEMORY_SCOPE_AGENT);
}

// acc[b, c] = ft_b[c%256] + fft_b[c%256]   (both halves get both biases)
__global__ void k_init_acc(float* __restrict__ acc,
                           const float* __restrict__ ft_b,
                           const float* __restrict__ fft_b) {
  int t = blockIdx.x * blockDim.x + threadIdx.x;
  if (t >= B_SZ * HID) return;
  int cc = t & (FT_OUT - 1);
  acc[t] = ft_b[cc] + fft_b[cc];
}

__global__ void k_zero(float* __restrict__ p, int n) {
  for (int t = blockIdx.x * blockDim.x + threadIdx.x; t < n;
       t += gridDim.x * blockDim.x)
    p[t] = 0.0f;
}

// ft_w [256, 40960] -> ft_wT [40960, 256], LDS-tiled 32x32
__global__ void k_transpose(const float* __restrict__ w, float* __restrict__ wT) {
  __shared__ float tile[32][33];
  int tx = threadIdx.x, ty = threadIdx.y;
  int i0 = blockIdx.x * 32;   // input col (FT_INN)
  int o0 = blockIdx.y * 32;   // input row (FT_OUT)
#pragma unroll
  for (int j = 0; j < 32; j += 8)
    tile[ty + j][tx] = w[(size_t)(o0 + ty + j) * FT_INN + (i0 + tx)];
  __syncthreads();
#pragma unroll
  for (int j = 0; j < 32; j += 8)
    wT[(size_t)(i0 + ty + j) * FT_OUT + (o0 + tx)] = tile[tx][ty + j];
}

// counts[side][row][col % 640] += values[k]
__global__ void k_hist(const int* __restrict__ stm, const int* __restrict__ nstm,
                       const float* __restrict__ vals, float* __restrict__ counts,
                       int nnz) {
  int t = blockIdx.x * blockDim.x + threadIdx.x;
  if (t >= 2 * nnz) return;
  int side = (t >= nnz) ? 1 : 0;
  int k = t - side * nnz;
  const int* idx = side ? nstm : stm;
  int row = idx[k];
  int col = idx[nnz + k] % VFT;
  atomAddF(counts + (size_t)side * B_SZ * VFT + (size_t)row * VFT + col, vals[k]);
}

// main scatter: acc[row, side*256 + c] += ft_wT[col, c] * val   (4 nnz per block)
__global__ void k_ft_scatter(const int* __restrict__ stm, const int* __restrict__ nstm,
                             const float* __restrict__ vals,
                             const float* __restrict__ ftT,
                             float* __restrict__ acc, int nnz) {
  int c = threadIdx.x;            // 0..255 output channel
  int base = blockIdx.x * 4;      // nnz index within [0, 2*nnz)
#pragma unroll
  for (int j = 0; j < 4; ++j) {
    int t = base + j;
    int side = (t >= nnz) ? 1 : 0;
    int k = t - side * nnz;
    const int* idx = side ? nstm : stm;
    int row = idx[k];
    int col = idx[nnz + k];
    float w = ftT[(size_t)col * FT_OUT + c] * vals[k];
    atomAddF(acc + (size_t)row * HID + side * FT_OUT + c, w);
  }
}

// acc[b, side*256+n] += sum_k counts[side][b][k] * fft_w[n][k] via V_WMMA_F32_16X16X4_F32
__global__ void k_fft_wmma(const float* __restrict__ counts,
                           const float* __restrict__ fft_w,
                           float* __restrict__ acc) {
  int wave = (blockIdx.x * blockDim.x + threadIdx.x) >> 5;
  int lane = threadIdx.x & 31;
  int side = wave >> 13;          // 8192 waves per side = 512 mtiles * 16 ntiles
  int rem  = wave & 8191;
  int mt   = rem >> 4;            // 0..511  (rows of B_SZ, 16 at a time)
  int nt   = rem & 15;            // 0..15   (cols of FT_OUT, 16 at a time)

  int mn    = lane & 15;          // A row / B col handled by this lane
  int khalf = (lane >> 4) << 1;   // 0 for lanes 0-15, 2 for lanes 16-31

  const float* pA = counts + (size_t)side * B_SZ * VFT +
                    (size_t)(mt * 16 + mn) * VFT + khalf;
  const float* pB = fft_w + (size_t)(nt * 16 + mn) * VFT + khalf;

  v8f c = {};
  for (int k = 0; k < VFT; k += 4) {
    v2f a = *(const v2f*)(pA + k);
    v2f b = *(const v2f*)(pB + k);
    c = __builtin_amdgcn_wmma_f32_16x16x4_f32(
        /*neg_a=*/false, a, /*neg_b=*/false, b,
        /*c_mod=*/(short)0, c, /*reuse_a=*/false, /*reuse_b=*/false);
  }

  // D layout: VGPR j -> M = (lane>=16 ? 8 : 0) + j, N = lane&15
  int mrow = (lane >> 4) << 3;
  float* pD = acc + (size_t)(mt * 16) * HID + side * FT_OUT + nt * 16 + mn;
#pragma unroll
  for (int j = 0; j < 8; ++j)
    atomAddF(pD + (size_t)(mrow + j) * HID, c[j]);
}

// out[b] = sigmoid( sum_c clip(acc[b,c],0,1) * out_w[c] + out_b )
__global__ void k_out(const float* __restrict__ acc,
                      const float* __restrict__ out_w,
                      const float* __restrict__ out_b,
                      float* __restrict__ out) {
  int wave = (blockIdx.x * blockDim.x + threadIdx.x) >> 5;
  int lane = threadIdx.x & 31;
  if (wave >= B_SZ) return;
  const float* row = acc + (size_t)wave * HID;
  float s = 0.0f;
#pragma unroll
  for (int j = 0; j < HID / 32; ++j) {
    int c = lane + j * 32;
    float h = row[c];
    h = fminf(fmaxf(h, 0.0f), 1.0f);
    s += h * out_w[c];
  }
#pragma unroll
  for (int off = 16; off; off >>= 1) s += __shfl_xor(s, off, 32);
  if (lane == 0) out[wave] = 1.0f / (1.0f + expf(-(s + out_b[0])));
}

extern "C" void kernel_launch(void* const* d_in, const int* in_sizes, int n_in,
                              void* d_out, int out_size, void* d_ws, size_t ws_size,
                              hipStream_t stream) {
  const int*   stm   = (const int*)d_in[0];
  const int*   nstm  = (const int*)d_in[1];
  const float* vals  = (const float*)d_in[2];
  // d_in[3] = size (B, fixed 8192 by setup)
  const float* ft_w  = (const float*)d_in[4];
  const float* ft_b  = (const float*)d_in[5];
  const float* fft_w = (const float*)d_in[6];
  const float* fft_b = (const float*)d_in[7];
  const float* out_w = (const float*)d_in[8];
  const float* out_b = (const float*)d_in[9];
  float* out = (float*)d_out;
  int nnz = in_sizes[0] / 2;  // 262144

  // workspace layout
  char* ws = (char*)d_ws;
  float* acc    = (float*)ws;                                   // B*512 f32 (16 MB)
  float* counts = (float*)(ws + (size_t)B_SZ * HID * 4);        // 2*B*640 f32 (40 MB)
  float* ftT    = (float*)(ws + (size_t)B_SZ * HID * 4
                              + (size_t)2 * B_SZ * VFT * 4);    // 40960*256 f32 (40 MB)

  k_init_acc<<<(B_SZ * HID) / 256, 256, 0, stream>>>(acc, ft_b, fft_b);
  k_zero<<<4096, 256, 0, stream>>>(counts, 2 * B_SZ * VFT);
  k_transpose<<<dim3(FT_INN / 32, FT_OUT / 32), dim3(32, 8), 0, stream>>>(ft_w, ftT);
  k_hist<<<(2 * nnz + 255) / 256, 256, 0, stream>>>(stm, nstm, vals, counts, nnz);
  k_ft_scatter<<<(2 * nnz) / 4, 256, 0, stream>>>(stm, nstm, vals, ftT, acc, nnz);
  k_fft_wmma<<<(16384 * 32) / 256, 256, 0, stream>>>(counts, fft_w, acc);
  k_out<<<(B_SZ * 32) / 256, 256, 0, stream>>>(acc, out_w, out_b, out);
}